// StructuredOrthgonalRandomFeatures_34187939676528
// MI455X (gfx1250) — compile-verified
//
#include <hip/hip_runtime.h>

typedef __attribute__((ext_vector_type(2))) float v2f;
typedef __attribute__((ext_vector_type(8))) float v8f;

#define RDIM    512   // R
#define NPCA    256   // P
#define NFEAT   8192  // F
#define SCHG    4     // S
#define MROWS   32    // M
#define AATOMS  64    // A
#define NSTACK  32    // NFEAT / NPCA
#define GROWS   (MROWS * AATOMS)   // 2048 fused (m,a) rows

// ---------------------------------------------------------------------------
// Phase 1: proj[s] = rep(2048x512) x reductors[s](512x256), full f32 WMMA.
// One wave32 per 16x16 output tile; K stepped 4 at a time via
// V_WMMA_F32_16X16X4_F32 (exact f32, no downcast needed).
//
// ISA VGPR layouts (CDNA5 ISA):
//   A (16x4 f32): lanes 0-15 -> rows, {vgpr0,vgpr1} = {K0,K1}; lanes 16-31 = {K2,K3}
//   B (4x16 f32): mirrored (lane half selects K pair, lane&15 selects column)
//   C/D (16x16 f32, 8 VGPRs): vgpr i -> row i (lanes 0-15) / row i+8 (lanes 16-31)
// ---------------------------------------------------------------------------
__global__ __launch_bounds__(32)
void sorf_proj_wmma(const float* __restrict__ rep,
                    const float* __restrict__ reductors,
                    float* __restrict__ proj)
{
    const int colTile = blockIdx.x;   // 0..15  (256/16)
    const int rowTile = blockIdx.y;   // 0..127 (2048/16)
    const int s       = blockIdx.z;   // 0..3
    const int lane    = threadIdx.x;  // wave32
    const int halfSel = lane >> 4;    // 0: K0/K1, 1: K2/K3
    const int lid     = lane & 15;

    const int row0 = rowTile * 16;
    const int col0 = colTile * 16;

    const float* Arow = rep + (size_t)(row0 + lid) * RDIM;
    const float* Bcol = reductors + (size_t)s * RDIM * NPCA + (col0 + lid);

    v8f acc = {};
    for (int k0 = 0; k0 < RDIM; k0 += 4) {
        const int kk = k0 + halfSel * 2;
        v2f a, b;
        a.x = Arow[kk];
        a.y = Arow[kk + 1];
        b.x = Bcol[(size_t)kk * NPCA];
        b.y = Bcol[(size_t)(kk + 1) * NPCA];
        // 8 args: (neg_a, A, neg_b, B, c_mod, C, reuse_a, reuse_b)
        acc = __builtin_amdgcn_wmma_f32_16x16x4_f32(
            false, a, false, b, (short)0, acc, false, false);
    }

    float* C = proj + (size_t)s * GROWS * NPCA;
#pragma unroll
    for (int i = 0; i < 8; ++i) {
        C[(size_t)(row0 + i + halfSel * 8) * NPCA + (col0 + lid)] = acc[i];
    }
}

// ---------------------------------------------------------------------------
// Phase 2: per (m,a) pair, pick s = charges[m,a]; for each of 32 stacks apply
// sign mask, 256-point FWHT in LDS (8 butterfly stages), cos(+bias), dot with
// alpha. One 256-thread block (8 waves) per pair; deterministic LDS reduction.
// Constants: fwht norm 256^-0.5 = 1/16; COEFF_NORM = sqrt(256)/16 = 1;
//            FEAT_NORM = sqrt(2/8192) = 1/64.
// ---------------------------------------------------------------------------
__global__ __launch_bounds__(256)
void sorf_features(const float* __restrict__ proj,
                   const int*   __restrict__ charges,
                   const float* __restrict__ Dmat,
                   const float* __restrict__ bias,
                   const float* __restrict__ alpha,
                   float* __restrict__ partial)
{
    __shared__ float buf[NPCA];
    __shared__ float red[256];

    const int g = blockIdx.x;     // fused (m,a) index
    const int t = threadIdx.x;    // 0..255 == PCA index
    const int s = charges[g];

    const float pv = proj[((size_t)s * GROWS + g) * NPCA + t];
    const float scale = 0.0625f;      // 256^-0.5 * COEFF_NORM

    const float* biasS = bias + (size_t)s * NFEAT;
    const float* Ds    = Dmat + (size_t)s * NSTACK * NPCA;

    float acc = 0.0f;
    for (int st = 0; st < NSTACK; ++st) {
        const float d = Ds[st * NPCA + t];
        buf[t] = (d >= 0.0f) ? pv : -pv;
        __syncthreads();
#pragma unroll
        for (int h = 1; h < NPCA; h <<= 1) {
            const float mine  = buf[t];
            const float other = buf[t ^ h];
            __syncthreads();
            buf[t] = (t & h) ? (other - mine) : (mine + other);
            __syncthreads();
        }
        const float x = buf[t] * scale;           // self-slot read; safe
        const int   f = st * NPCA + t;
        acc += cosf(x + biasS[f]) * alpha[f];
    }

    red[t] = acc;
    __syncthreads();
#pragma unroll
    for (int w = 128; w > 0; w >>= 1) {
        if (t < w) red[t] += red[t + w];
        __syncthreads();
    }
    if (t == 0) partial[g] = red[0] * 0.015625f;  // FEAT_NORM = 1/64
}

// ---------------------------------------------------------------------------
// Phase 3: out[m] = sum over 64 atoms of partial[m*64+a] (deterministic).
// ---------------------------------------------------------------------------
__global__ __launch_bounds__(64)
void sorf_reduce(const float* __restrict__ partial, float* __restrict__ out)
{
    __shared__ float red[AATOMS];
    const int m = blockIdx.x;
    const int t = threadIdx.x;
    red[t] = partial[m * AATOMS + t];
    __syncthreads();
#pragma unroll
    for (int w = AATOMS / 2; w > 0; w >>= 1) {
        if (t < w) red[t] += red[t + w];
        __syncthreads();
    }
    if (t == 0) out[m] = red[0];
}

extern "C" void kernel_launch(void* const* d_in, const int* in_sizes, int n_in,
                              void* d_out, int out_size, void* d_ws, size_t ws_size,
                              hipStream_t stream)
{
    const float* rep       = (const float*)d_in[0];  // (32,64,512)
    const int*   charges   = (const int*)  d_in[1];  // (32,64)
    const float* reductors = (const float*)d_in[2];  // (4,512,256)
    const float* Dmat      = (const float*)d_in[3];  // (4,1,32,256)
    const float* bias      = (const float*)d_in[4];  // (4,8192)
    const float* alpha     = (const float*)d_in[5];  // (8192,)
    float* out = (float*)d_out;                      // (32,)

    float* proj    = (float*)d_ws;                        // 4*2048*256 f32 = 8 MB
    float* partial = proj + (size_t)SCHG * GROWS * NPCA;  // 2048 f32

    dim3 g1(NPCA / 16, GROWS / 16, SCHG);
    sorf_proj_wmma<<<g1, 32, 0, stream>>>(rep, reductors, proj);
    sorf_features<<<GROWS, 256, 0, stream>>>(proj, charges, Dmat, bias, alpha, partial);
    sorf_reduce<<<MROWS, AATOMS, 0, stream>>>(partial, out);
}